// CRF_73873437491653
// MI455X (gfx1250) — compile-verified
//
#include <hip/hip_runtime.h>
#include <hip/hip_bf16.h>

typedef float v2f __attribute__((ext_vector_type(2)));
typedef float v4f __attribute__((ext_vector_type(4)));
typedef float v8f __attribute__((ext_vector_type(8)));
typedef unsigned int u32x4 __attribute__((ext_vector_type(4)));
typedef int i32x4 __attribute__((ext_vector_type(4)));
typedef int i32x8 __attribute__((ext_vector_type(8)));

#define CRF_B  256
#define CRF_T  128
#define CRF_HU 512
#define CRF_C  64
#define K3     (3 * CRF_HU)   // 1536
#define LROW   516            // padded LDS row stride (words) -> conflict-free
#define MBLK   64             // M rows (time steps) per workgroup

#if defined(__has_builtin)
#if __has_builtin(__builtin_amdgcn_tensor_load_to_lds) && \
    __has_builtin(__builtin_amdgcn_s_wait_tensorcnt)
#define CRF_HAVE_TDM 1
#endif
#endif

// ---------------------------------------------------------------------------
// Kernel 1: scores[b,t,c] = sum_k context[b,t,k] * W[c,k] + bias[c]
// context = [H[t-1] | H[t] | H[t+1]]  (zeros outside [0,T))
// Grid = B * (T/64) workgroups, 128 threads (4 waves).
// Each wave: N-tile of 16 classes, 4 M-subtiles of 16 rows, f32 WMMA K=4.
// H tile (t0-1 .. t0+64, 66 rows) staged to LDS by the Tensor Data Mover with
// a 4-DWORD pad every 128 DWORDs (row stride 516 words, bank-conflict-free).
// ---------------------------------------------------------------------------
__global__ __launch_bounds__(128) void crf_scores_wmma(
    const float* __restrict__ H, const float* __restrict__ W,
    const float* __restrict__ bias, float* __restrict__ scores) {
  __shared__ float hst[(MBLK + 2) * LROW];  // 66 rows * 516 words = 136224 B

  const int tid  = threadIdx.x;
  const int wave = tid >> 5;
  const int lane = tid & 31;
  const int b    = blockIdx.x >> 1;
  const int t0   = (blockIdx.x & 1) * MBLK;

#if CRF_HAVE_TDM
  {
    int t_start = t0 - 1;
    int row0    = 0;
    if (t_start < 0) { t_start = 0; row0 = 1; }
    if (row0) {  // zero the leading ghost row (t = -1)
      for (int i = tid; i < CRF_HU; i += 128) hst[i] = 0.f;
    }
    if (tid < 32) {  // one wave issues the TDM descriptor
      const unsigned long long ga =
          (unsigned long long)(const void*)(H + ((size_t)b * CRF_T + t_start) * CRF_HU);
      const unsigned int ldsa = (unsigned int)(__SIZE_TYPE__)(&hst[row0 * LROW]);
      const int tile_rows  = (MBLK + 2) - row0;
      const int valid_rows = CRF_T - t_start;  // TDM zero-fills rows past this

      u32x4 g0;
      g0.x = 1u;                                   // count=1, no gather
      g0.y = ldsa;                                 // lds_addr [63:32]
      g0.z = (unsigned int)ga;                     // global_addr [95:64]
      g0.w = (unsigned int)(ga >> 32) | (2u << 30);// addr[56:32] | type=2

      i32x8 g1;
      g1[0] = (int)((2u << 16)    // data_size = 4B
                    | (1u << 20)  // pad_enable
                    | (6u << 22)  // pad_interval: every 128 DWORDs
                    | (3u << 25));// pad_amount: 4 DWORDs -> stride 516 words
      g1[1] = (int)((unsigned)CRF_HU << 16);          // tensor_dim0 = 512
      g1[2] = (int)((unsigned)valid_rows << 16);      // tensor_dim1
      g1[3] = (int)((unsigned)CRF_HU << 16);          // tile_dim0 = 512
      g1[4] = tile_rows;                              // tile_dim1 (tile_dim2=0)
      g1[5] = CRF_HU;                                 // tensor_dim0_stride=512
      g1[6] = 0;
      g1[7] = 0;

      i32x4 z4 = {0, 0, 0, 0};
#if defined(__clang_major__) && __clang_major__ >= 23
      i32x8 z8 = {0, 0, 0, 0, 0, 0, 0, 0};
      __builtin_amdgcn_tensor_load_to_lds(g0, g1, z4, z4, z8, 0);
#else
      __builtin_amdgcn_tensor_load_to_lds(g0, g1, z4, z4, 0);
#endif
      __builtin_amdgcn_s_wait_tensorcnt(0);
    }
  }
  __syncthreads();
#else
  // Fallback: manual float4 staging, zero-fill out-of-range rows.
  for (int idx = tid; idx < (MBLK + 2) * (CRF_HU / 4); idx += 128) {
    const int row = idx >> 7;  // HU/4 == 128
    const int c4  = idx & 127;
    const int t   = t0 + row - 1;
    v4f val = {0.f, 0.f, 0.f, 0.f};
    if (t >= 0 && t < CRF_T) {
      const v4f* src = (const v4f*)(H + (size_t)(b * CRF_T + t) * CRF_HU);
      val = src[c4];
    }
    *(v4f*)(&hst[row * LROW + c4 * 4]) = val;
  }
  __syncthreads();
#endif

  const int half = lane >> 4;          // 0: K={0,1}, 1: K={2,3}
  const int l16  = lane & 15;
  const int ccol = wave * 16 + l16;    // this lane's class column
  const float* wbase = W + (size_t)ccol * K3 + 2 * half;

  v8f acc0 = {};
  v8f acc1 = {};
  v8f acc2 = {};
  v8f acc3 = {};

#pragma unroll 1
  for (int region = 0; region < 3; ++region) {  // 0=prev, 1=cur, 2=next
    const float* wr = wbase + region * CRF_HU;
    const float* a0 = &hst[(l16 + region) * LROW + 2 * half];
    const float* a2 = a0 + 32 * LROW;
#pragma unroll 2
    for (int kk = 0; kk < CRF_HU; kk += 4) {
      v2f bf = *(const v2f*)(wr + kk);
      v2f f0 = *(const v2f*)(a0 + kk);
      v2f f1 = *(const v2f*)(a0 + 16 * LROW + kk);
      v2f f2 = *(const v2f*)(a2 + kk);
      v2f f3 = *(const v2f*)(a2 + 16 * LROW + kk);
      acc0 = __builtin_amdgcn_wmma_f32_16x16x4_f32(false, f0, false, bf,
                                                   (short)0, acc0, false, false);
      acc1 = __builtin_amdgcn_wmma_f32_16x16x4_f32(false, f1, false, bf,
                                                   (short)0, acc1, false, false);
      acc2 = __builtin_amdgcn_wmma_f32_16x16x4_f32(false, f2, false, bf,
                                                   (short)0, acc2, false, false);
      acc3 = __builtin_amdgcn_wmma_f32_16x16x4_f32(false, f3, false, bf,
                                                   (short)0, acc3, false, false);
    }
  }

  const float bv = bias[ccol];
  float* out0 = scores + (size_t)(b * CRF_T + t0) * CRF_C;
#pragma unroll
  for (int r = 0; r < 8; ++r) {
    const int m0 = r + 8 * half;  // C/D layout: vgpr r, lane half -> M
    out0[m0 * CRF_C + ccol]        = acc0[r] + bv;
    out0[(16 + m0) * CRF_C + ccol] = acc1[r] + bv;
    out0[(32 + m0) * CRF_C + ccol] = acc2[r] + bv;
    out0[(48 + m0) * CRF_C + ccol] = acc3[r] + bv;
  }
}

// ---------------------------------------------------------------------------
// Kernel 2: per-batch forward recursion (logZ) + numerator. 64 threads/block.
// new_j = em_j + m + log( sum_i exp(lp_i - m) * E[i][j] ),  E = exp(trans)
// ---------------------------------------------------------------------------
__global__ __launch_bounds__(64) void crf_scan(
    const float* __restrict__ scores, const float* __restrict__ startT,
    const float* __restrict__ endT, const float* __restrict__ trans,
    const int* __restrict__ y, const float* __restrict__ ymask,
    const int* __restrict__ ylen, float* __restrict__ out) {
  __shared__ float E[CRF_C * CRF_C];
  __shared__ float lp[CRF_C];
  __shared__ float pv[CRF_C];
  __shared__ float red[CRF_C];

  const int b = blockIdx.x;
  const int j = threadIdx.x;
  const float* sc = scores + (size_t)b * CRF_T * CRF_C;

  for (int i = 0; i < CRF_C; ++i) E[i * CRF_C + j] = __expf(trans[i * CRF_C + j]);

  lp[j] = startT[j] + sc[j];
  const int len = ylen[b];
  __syncthreads();

  for (int t = 1; t < CRF_T; ++t) {
    red[j] = lp[j];
    __syncthreads();
    for (int off = 32; off > 0; off >>= 1) {
      if (j < off) red[j] = fmaxf(red[j], red[j + off]);
      __syncthreads();
    }
    const float m = red[0];
    pv[j] = __expf(lp[j] - m);
    __syncthreads();

    float s = 0.f;
#pragma unroll 8
    for (int i = 0; i < CRF_C; ++i) s += pv[i] * E[i * CRF_C + j];

    const float nv = sc[t * CRF_C + j] + m + __logf(s);
    if (t < len) lp[j] = nv;  // uniform branch: mask[b,t] == (t < len)
    __syncthreads();
  }

  // logZ = logsumexp_j(lp_j + end_j)
  red[j] = lp[j] + endT[j];
  __syncthreads();
  for (int off = 32; off > 0; off >>= 1) {
    if (j < off) red[j] = fmaxf(red[j], red[j + off]);
    __syncthreads();
  }
  const float m2 = red[0];
  pv[j] = __expf(lp[j] + endT[j] - m2);
  __syncthreads();
  for (int off = 32; off > 0; off >>= 1) {
    if (j < off) pv[j] += pv[j + off];
    __syncthreads();
  }
  const float logZ = m2 + __logf(pv[0]);

  // numerator: gold-path emissions + transitions, reduced over t
  float partial = 0.f;
  for (int t = j; t < CRF_T; t += CRF_C) {
    const float mk = ymask[b * CRF_T + t];
    const int   yt = y[b * CRF_T + t];
    if (t <= CRF_T - 2) partial += mk * sc[t * CRF_C + yt];
    if (t >= 1)         partial += mk * trans[y[b * CRF_T + t - 1] * CRF_C + yt];
  }
  red[j] = partial;
  __syncthreads();
  for (int off = 32; off > 0; off >>= 1) {
    if (j < off) red[j] += red[j + off];
    __syncthreads();
  }

  if (j == 0) {
    const int last = y[b * CRF_T + (len - 1)];
    float num = startT[y[b * CRF_T]] + red[0] + endT[last] +
                sc[(CRF_T - 1) * CRF_C + last] * ymask[b * CRF_T + CRF_T - 1];
    out[b] = num - logZ;
  }
}

// ---------------------------------------------------------------------------
extern "C" void kernel_launch(void* const* d_in, const int* in_sizes, int n_in,
                              void* d_out, int out_size, void* d_ws, size_t ws_size,
                              hipStream_t stream) {
  const float* H     = (const float*)d_in[0];
  const float* W     = (const float*)d_in[1];
  const float* bias  = (const float*)d_in[2];
  const float* st    = (const float*)d_in[3];
  const float* en    = (const float*)d_in[4];
  const float* trans = (const float*)d_in[5];
  const int*   y     = (const int*)d_in[6];
  const float* ymask = (const float*)d_in[7];
  const int*   ylen  = (const int*)d_in[8];

  float* scores = (float*)d_ws;  // B*T*C fp32 = 8 MB scratch

  crf_scores_wmma<<<CRF_B * (CRF_T / MBLK), 128, 0, stream>>>(H, W, bias, scores);
  crf_scan<<<CRF_B, 64, 0, stream>>>(scores, st, en, trans, y, ymask, ylen,
                                     (float*)d_out);
}